// re_unit_45853070852709
// MI455X (gfx1250) — compile-verified
//
#include <hip/hip_runtime.h>
#include <hip/hip_bf16.h>

#define Bsz 8
#define Lsz 1024
#define Hsz 300
#define Ksz 5
#define PD 50
#define NC 11234
#define Fsz 650
#define LN_EPS 1e-5f

typedef __attribute__((ext_vector_type(2))) float v2f;
typedef __attribute__((ext_vector_type(8))) float v8f;

// ---------------------------------------------------------------------------
// Kernel 1: R[r][h] = b1[h] + sum_c RPE[r][c] * W1[h][600+c]
//   RPE[r][c] = sum_{r'} (L - |r'-K|) * exp(-(r-r')^2) * pos_W[r'][c]
// 11 blocks (one per rel value), 320 threads.
// ---------------------------------------------------------------------------
__global__ void build_R_kernel(const float* __restrict__ pos_W,
                               const float* __restrict__ W1,
                               const float* __restrict__ b1,
                               float* __restrict__ R) {
    __shared__ float rpe[PD];
    const int r = blockIdx.x;     // 0..10
    const int t = threadIdx.x;    // 0..319
    if (t < PD) {
        float s = 0.f;
        #pragma unroll
        for (int rp = 0; rp < 2 * Ksz + 1; ++rp) {
            int arel = rp > Ksz ? rp - Ksz : Ksz - rp;
            float cnt = (float)(Lsz - arel);
            float d = (float)(r - rp);
            s += cnt * __expf(-d * d) * pos_W[rp * PD + t];
        }
        rpe[t] = s;
    }
    __syncthreads();
    if (t < Hsz) {
        float s = b1[t];
        const float* w = W1 + (size_t)t * Fsz + 2 * Hsz;   // columns 600..649
        #pragma unroll 10
        for (int c = 0; c < PD; ++c) s += rpe[c] * w[c];
        R[r * Hsz + t] = s;
    }
}

// ---------------------------------------------------------------------------
// Kernel 2: band index tables + emo_cau_pos output (1-based, as floats).
// One thread per row i; closed-form prefix into the band ordering.
// ---------------------------------------------------------------------------
__global__ void build_idx_kernel(int* __restrict__ ii, int* __restrict__ jj,
                                 float* __restrict__ pos_out) {
    int i = blockIdx.x * blockDim.x + threadIdx.x;
    if (i >= Lsz) return;
    int pfx;
    if (i <= Ksz)              pfx = i * (i + 11) / 2;                       // rows 0..4
    else if (i <= Lsz - Ksz)   pfx = 40 + (i - Ksz) * (2 * Ksz + 1);         // full rows of 11
    else                       pfx = 11194 + ((10 + (1030 - i)) * (i - 1019)) / 2;
    int j0 = i - Ksz < 0 ? 0 : i - Ksz;
    int j1 = i + Ksz > Lsz - 1 ? Lsz - 1 : i + Ksz;
    int p = pfx;
    for (int j = j0; j <= j1; ++j, ++p) {
        ii[p] = i;
        jj[p] = j;
        pos_out[2 * p]     = (float)(i + 1);
        pos_out[2 * p + 1] = (float)(j + 1);
    }
}

// ---------------------------------------------------------------------------
// Kernel 3: fp32 WMMA GEMM   dst[m][n] = sum_k (src[m][k]+hsh[m][k]) * W1[n][koff+k]
// M=8192, N=300, K=300. One 16x16 tile per wave via V_WMMA_F32_16X16X4_F32.
// 9728 tiles = 1216 blocks * 8 waves (exact).
// B out-of-range lanes: pointer clamped to a valid row, value zeroed by a
// 0/1 float mask -> fully unconditional b64 loads, no EXEC toggling in loop.
// ---------------------------------------------------------------------------
__global__ void gemm_wmma_kernel(const float* __restrict__ src,
                                 const float* __restrict__ hsh,
                                 const float* __restrict__ W1,
                                 int koff,
                                 float* __restrict__ dst) {
    const int NT = 19;                                   // ceil(300/16) N-tiles
    const int wave = (blockIdx.x * blockDim.x + threadIdx.x) >> 5;
    const int lane = threadIdx.x & 31;
    const int mt = wave / NT;
    const int nt = wave % NT;
    const int m0 = mt * 16;
    const int n0 = nt * 16;
    const int half = lane >> 4;                          // 0: K={0,1}, 1: K={2,3}
    const int lr   = lane & 15;

    // A fragment: lane lr holds row m0+lr, columns k+2*half, k+2*half+1
    const float* srow  = src + (size_t)(m0 + lr) * Hsz + 2 * half;
    const float* shrow = hsh + (size_t)(m0 + lr) * Hsz + 2 * half;
    // B fragment: lane lr holds column n0+lr (= W1 row), rows k+2*half(+1)
    const int   ncol  = n0 + lr;
    const float nmask = (ncol < Hsz) ? 1.0f : 0.0f;      // zero pad column
    const float* wrow = W1 + (size_t)(ncol < Hsz ? ncol : 0) * Fsz + koff + 2 * half;

    v8f c = {0.f, 0.f, 0.f, 0.f, 0.f, 0.f, 0.f, 0.f};
    for (int k = 0; k < Hsz; k += 4) {                   // 75 WMMA issues
        v2f a, b;
        a.x = srow[k]     + shrow[k];
        a.y = srow[k + 1] + shrow[k + 1];
        b.x = wrow[k]     * nmask;                       // unconditional b64 load
        b.y = wrow[k + 1] * nmask;
        c = __builtin_amdgcn_wmma_f32_16x16x4_f32(
                /*neg_a=*/false, a, /*neg_b=*/false, b,
                /*c_mod=*/(short)0, c, /*reuse_a=*/false, /*reuse_b=*/false);
    }
    if (ncol < Hsz) {
        const int mrow = m0 + 8 * half;                  // C layout: v -> M = v + 8*half
        #pragma unroll
        for (int v = 0; v < 8; ++v)
            dst[(size_t)(mrow + v) * Hsz + ncol] = c[v];
    }
}

// ---------------------------------------------------------------------------
// Kernel 4: per-pair fuse: h = A[b,i] + Cc[b,j] + R[rel]; LayerNorm; ELU;
//           out = h . W2 + b2.   One wave32 per (b, pair).
// 89872 waves = 11234 blocks * 8 waves (exact).
// Elements 0..287 handled unconditionally (9 per lane); single guarded tail.
// ---------------------------------------------------------------------------
__global__ void pair_ln_out_kernel(const float* __restrict__ A,
                                   const float* __restrict__ Cc,
                                   const float* __restrict__ R,
                                   const int* __restrict__ ii,
                                   const int* __restrict__ jj,
                                   const float* __restrict__ ln_g,
                                   const float* __restrict__ ln_b,
                                   const float* __restrict__ W2,
                                   const float* __restrict__ b2,
                                   float* __restrict__ out) {
    const int wave = (blockIdx.x * blockDim.x + threadIdx.x) >> 5;
    const int lane = threadIdx.x & 31;
    const int b = wave / NC;
    const int p = wave % NC;
    const int i = ii[p];
    const int j = jj[p];
    const int rel = j - i + Ksz;

    const float* Ar = A  + (size_t)(b * Lsz + i) * Hsz;
    const float* Cr = Cc + (size_t)(b * Lsz + j) * Hsz;
    const float* Rr = R  + (size_t)rel * Hsz;

    const int  htail = lane + 9 * 32;                    // 288..319
    const bool tailv = htail < Hsz;                      // lanes 0..11

    float x[10];
    float s = 0.f;
    #pragma unroll
    for (int q = 0; q < 9; ++q) {                        // h <= 287 < 300: no guard
        const int h = lane + q * 32;
        x[q] = Ar[h] + Cr[h] + Rr[h];
        s += x[q];
    }
    x[9] = tailv ? (Ar[htail] + Cr[htail] + Rr[htail]) : 0.f;
    s += x[9];

    #pragma unroll
    for (int off = 16; off > 0; off >>= 1) s += __shfl_xor(s, off, 32);
    const float mu = s * (1.0f / (float)Hsz);

    float sq = 0.f;
    #pragma unroll
    for (int q = 0; q < 9; ++q) { const float d = x[q] - mu; sq += d * d; }
    if (tailv) { const float d = x[9] - mu; sq += d * d; }
    #pragma unroll
    for (int off = 16; off > 0; off >>= 1) sq += __shfl_xor(sq, off, 32);
    const float inv = rsqrtf(sq * (1.0f / (float)Hsz) + LN_EPS);

    float acc = 0.f;
    #pragma unroll
    for (int q = 0; q < 9; ++q) {
        const int h = lane + q * 32;
        float y = (x[q] - mu) * inv * ln_g[h] + ln_b[h];
        float e = y > 0.f ? y : expm1f(y);               // ELU(alpha=1)
        acc += e * W2[h];
    }
    if (tailv) {
        float y = (x[9] - mu) * inv * ln_g[htail] + ln_b[htail];
        float e = y > 0.f ? y : expm1f(y);
        acc += e * W2[htail];
    }
    #pragma unroll
    for (int off = 16; off > 0; off >>= 1) acc += __shfl_xor(acc, off, 32);
    if (lane == 0) out[(size_t)b * NC + p] = acc + b2[0];
}

// ---------------------------------------------------------------------------
extern "C" void kernel_launch(void* const* d_in, const int* in_sizes, int n_in,
                              void* d_out, int out_size, void* d_ws, size_t ws_size,
                              hipStream_t stream) {
    const float* h_e     = (const float*)d_in[0];
    const float* h_c     = (const float*)d_in[1];
    const float* h_share = (const float*)d_in[2];
    // d_in[3] = mask (unused)
    const float* pos_W   = (const float*)d_in[4];
    const float* W1      = (const float*)d_in[5];
    const float* b1      = (const float*)d_in[6];
    const float* ln_g    = (const float*)d_in[7];
    const float* ln_b    = (const float*)d_in[8];
    const float* W2      = (const float*)d_in[9];
    const float* b2      = (const float*)d_in[10];

    float* out     = (float*)d_out;             // [B*NC] scores
    float* pos_out = out + (size_t)Bsz * NC;    // [NC*2] 1-based (i,j) as floats

    // Workspace layout (~19.8 MB)
    float* A  = (float*)d_ws;                   // [8192*300]
    float* Cc = A  + (size_t)Bsz * Lsz * Hsz;   // [8192*300]
    float* R  = Cc + (size_t)Bsz * Lsz * Hsz;   // [11*300]
    int*   ii = (int*)(R + (2 * Ksz + 1) * Hsz);
    int*   jj = ii + NC;

    // 1) tiny: 11 distinct rel_pos_emb rows folded through W1's last 50 cols + b1
    build_R_kernel<<<2 * Ksz + 1, 320, 0, stream>>>(pos_W, W1, b1, R);

    // 2) band index tables + emo_cau_pos output
    build_idx_kernel<<<Lsz / 256, 256, 0, stream>>>(ii, jj, pos_out);

    // 3) A = (h_e + h_share) @ W1[:, 0:300]^T ; Cc = (h_c + h_share) @ W1[:, 300:600]^T
    //    512*19 = 9728 wave-tiles -> 1216 blocks of 8 waves
    gemm_wmma_kernel<<<1216, 256, 0, stream>>>(h_e, h_share, W1, 0,   A);
    gemm_wmma_kernel<<<1216, 256, 0, stream>>>(h_c, h_share, W1, Hsz, Cc);

    // 4) per-pair gather-add + LayerNorm + ELU + dot(W2) ; 89872 waves
    pair_ln_out_kernel<<<(Bsz * NC) / 8, 256, 0, stream>>>(
        A, Cc, R, ii, jj, ln_g, ln_b, W2, b2, out);
}